// CoarseMatching_35064113005039
// MI455X (gfx1250) — compile-verified
//
#include <hip/hip_runtime.h>
#include <math.h>

#define NREF 16384
#define MSRC 16384
#define DDIM 256
#define MBLK 128      // ref rows per block
#define NBLK 64       // src cols per block
#define KCH  32       // K chunk staged in LDS (dwords per row)
#define LDP  36       // padded LDS row stride (floats): TDM pads 32dw + 4dw
#define NBINS 4096
#define CAP  16384    // candidate buffer capacity
#define NCHUNK (DDIM / KCH)
#define A_ELEMS (MBLK * LDP)   // 4608 floats per A buffer
#define B_ELEMS (NBLK * LDP)   // 2304 floats per B buffer

typedef __attribute__((ext_vector_type(2))) float    v2f;
typedef __attribute__((ext_vector_type(8))) float    v8f;
typedef __attribute__((ext_vector_type(4))) unsigned v4u;
typedef __attribute__((ext_vector_type(8))) int      v8i;
typedef __attribute__((ext_vector_type(4))) int      v4i;

// Order-preserving map float -> uint32 (monotone increasing)
__device__ __forceinline__ unsigned fkey(float x) {
  unsigned u = __float_as_uint(x);
  return (u & 0x80000000u) ? ~u : (u | 0x80000000u);
}

// Issue one TDM load: tile (tile_rows x KCH dwords) from a row-major [rows x DDIM]
// f32 tensor at (row0, kb) into LDS at ldsOff, padding each 32-dword row to 36 dwords.
__device__ __forceinline__ void tdm_load_tile(const float* base, unsigned row0,
                                              unsigned kb, unsigned tile_rows,
                                              unsigned ldsOff)
{
  unsigned long long ga = (unsigned long long)(uintptr_t)base
                        + ((unsigned long long)row0 * DDIM + kb) * 4ull;
  v4u g0;
  g0.x = 1u;                                           // count=1 (valid descriptor)
  g0.y = ldsOff;                                       // LDS byte address
  g0.z = (unsigned)ga;                                 // global_addr[31:0]
  g0.w = ((unsigned)(ga >> 32) & 0x01FFFFFFu) | (2u << 30);  // addr[56:32] | type=2

  // w0: data_size=4B(2<<16) | pad_enable(1<<20) | pad_interval=32dw(4<<22) | pad_amount=4dw(3<<25)
  v8i g1 = {
    (int)((2u << 16) | (1u << 20) | (4u << 22) | (3u << 25)),
    (int)(((unsigned)DDIM & 0xFFFFu) << 16),           // tensor_dim0 low16 @ bits 63:48
    (int)(((unsigned)NREF & 0xFFFFu) << 16),           // dim0 hi16=0 | tensor_dim1 low16
    (int)((unsigned)KCH << 16),                        // dim1 hi16=0 | tile_dim0=32
    (int)tile_rows,                                    // tile_dim1 | tile_dim2=0
    (int)DDIM,                                         // tensor_dim0_stride low32
    0, 0
  };
  v4i g2 = {0, 0, 0, 0};
  v4i g3 = {0, 0, 0, 0};
  v8i g4 = {0, 0, 0, 0, 0, 0, 0, 0};
  __builtin_amdgcn_tensor_load_to_lds(g0, g1, g2, g3, g4, 0);
}

// Fused GEMM + (mode 0: histogram of S-keys) / (mode 1: collect candidates >= tau)
__global__ __launch_bounds__(256) void gemm_topk_pass(
    const float* __restrict__ ref, const float* __restrict__ src,
    unsigned* ghist, const unsigned* ctrl, unsigned* cnt,
    float* cscore, unsigned* cidx, int mode)
{
  // A0 | A1 | B0 | B1 ; the 16KB histogram aliases the front (used only post-loop)
  __shared__ __align__(16) float smem[2 * A_ELEMS + 2 * B_ELEMS];
  float* Ab = smem;
  float* Bb = smem + 2 * A_ELEMS;
  const unsigned smemBase = (unsigned)(unsigned long long)(uintptr_t)(void*)smem;
  const unsigned ldsA0 = smemBase;
  const unsigned ldsB0 = smemBase + 2u * A_ELEMS * 4u;

  const int tid  = threadIdx.x;
  const int lane = tid & 31;
  const int wid  = tid >> 5;
  const int wm   = wid >> 1;              // 0..3 : wave row in block tile
  const int wn   = wid & 1;               // 0..1 : wave col in block tile
  const int r    = lane & 15;
  const int koff = (lane >> 4) << 1;      // 0 or 2 (per WMMA f32 A/B layout)

  const int row0 = blockIdx.y * MBLK;
  const int col0 = blockIdx.x * NBLK;

  v8f acc[2][2];
  acc[0][0] = (v8f){}; acc[0][1] = (v8f){};
  acc[1][0] = (v8f){}; acc[1][1] = (v8f){};

  // Prologue: chunk 0 into buffer 0 (wave 0 issues the DMAs; EXEC ignored by TDM)
  if (wid == 0) {
    tdm_load_tile(ref, (unsigned)row0, 0u, MBLK, ldsA0);
    tdm_load_tile(src, (unsigned)col0, 0u, NBLK, ldsB0);
  }

  for (int ch = 0; ch < NCHUNK; ++ch) {
    const int cur = ch & 1;
    if (wid == 0) {
      if (ch + 1 < NCHUNK) {
        const int nxt = cur ^ 1;
        tdm_load_tile(ref, (unsigned)row0, (unsigned)((ch + 1) * KCH), MBLK,
                      ldsA0 + (unsigned)nxt * A_ELEMS * 4u);
        tdm_load_tile(src, (unsigned)col0, (unsigned)((ch + 1) * KCH), NBLK,
                      ldsB0 + (unsigned)nxt * B_ELEMS * 4u);
        __builtin_amdgcn_s_wait_tensorcnt(2);  // retire current chunk's 2 loads
      } else {
        __builtin_amdgcn_s_wait_tensorcnt(0);
      }
    }
    __syncthreads();  // publish LDS tile to all waves

    const float* Ac = Ab + cur * A_ELEMS;
    const float* Bc = Bb + cur * B_ELEMS;
    #pragma unroll
    for (int kk = 0; kk < KCH; kk += 4) {
      v2f a0 = *(const v2f*)(Ac + (wm * 32 +  0 + r) * LDP + kk + koff);
      v2f a1 = *(const v2f*)(Ac + (wm * 32 + 16 + r) * LDP + kk + koff);
      v2f b0 = *(const v2f*)(Bc + (wn * 32 +  0 + r) * LDP + kk + koff);
      v2f b1 = *(const v2f*)(Bc + (wn * 32 + 16 + r) * LDP + kk + koff);
      acc[0][0] = __builtin_amdgcn_wmma_f32_16x16x4_f32(false, a0, false, b0, (short)0, acc[0][0], false, false);
      acc[0][1] = __builtin_amdgcn_wmma_f32_16x16x4_f32(false, a0, false, b1, (short)0, acc[0][1], false, false);
      acc[1][0] = __builtin_amdgcn_wmma_f32_16x16x4_f32(false, a1, false, b0, (short)0, acc[1][0], false, false);
      acc[1][1] = __builtin_amdgcn_wmma_f32_16x16x4_f32(false, a1, false, b1, (short)0, acc[1][1], false, false);
    }
    __syncthreads();  // all reads of buffer `cur` done before it is refilled
  }

  // Epilogue: C layout — VGPR e: lanes 0-15 -> M=e, lanes 16-31 -> M=e+8; N = lane&15
  const int mbase = row0 + wm * 32 + ((lane >> 4) << 3);
  const int nbase = col0 + wn * 32 + (lane & 15);

  if (mode == 0) {
    unsigned* shist = (unsigned*)smem;   // alias tile buffers (done with them)
    for (int i = tid; i < NBINS; i += 256) shist[i] = 0u;
    __syncthreads();
    #pragma unroll
    for (int fm = 0; fm < 2; ++fm) {
      #pragma unroll
      for (int fn = 0; fn < 2; ++fn) {
        v8f f = acc[fm][fn];
        #pragma unroll
        for (int e = 0; e < 8; ++e) {
          atomicAdd(&shist[fkey(f[e]) >> 20], 1u);
        }
      }
    }
    __syncthreads();
    for (int i = tid; i < NBINS; i += 256) {
      unsigned v = shist[i];
      if (v) atomicAdd(ghist + i, v);
    }
  } else {
    const unsigned tau = ctrl[0];
    #pragma unroll
    for (int fm = 0; fm < 2; ++fm) {
      #pragma unroll
      for (int fn = 0; fn < 2; ++fn) {
        v8f f = acc[fm][fn];
        #pragma unroll
        for (int e = 0; e < 8; ++e) {
          float sv = f[e];
          if (fkey(sv) >= tau) {
            unsigned p = atomicAdd(cnt, 1u);
            if (p < CAP) {
              cscore[p] = expf(2.0f * sv - 2.0f);
              cidx[p]   = (unsigned)(mbase + fm * 16 + e) * (unsigned)MSRC
                        + (unsigned)(nbase + fn * 16);
            }
          }
        }
      }
    }
  }
}

// Scan histogram top-down for the bin containing rank 256; tau = its lower edge.
__global__ void find_thresh(const unsigned* ghist, unsigned* ctrl)
{
  if (threadIdx.x == 0) {
    unsigned cum = 0;
    int b = NBINS - 1;
    for (; b >= 0; --b) {
      cum += ghist[b];
      if (cum >= 256u) break;
    }
    if (b < 0) b = 0;
    ctrl[0] = ((unsigned)b) << 20;
  }
}

// Exact top-256 selection over <= CAP candidates: key (score desc, index asc),
// matching jax.lax.top_k tie behavior. Deterministic regardless of fill order.
__global__ __launch_bounds__(256) void final_select(
    unsigned* ctrl, float* cscore, unsigned* cidx, float* out)
{
  __shared__ float    ss[256];
  __shared__ unsigned si[256];
  __shared__ unsigned sp[256];
  const int tid = threadIdx.x;
  unsigned count = ctrl[1];
  if (count > CAP) count = CAP;

  for (int sel = 0; sel < 256; ++sel) {
    float    bs = -INFINITY;
    unsigned bi = 0xFFFFFFFFu;
    unsigned bp = 0xFFFFFFFFu;
    for (unsigned i = tid; i < count; i += 256) {
      float s = cscore[i];
      unsigned id = cidx[i];
      if (s > bs || (s == bs && id < bi)) { bs = s; bi = id; bp = i; }
    }
    ss[tid] = bs; si[tid] = bi; sp[tid] = bp;
    __syncthreads();
    for (int off = 128; off > 0; off >>= 1) {
      if (tid < off) {
        float s2 = ss[tid + off];
        unsigned i2 = si[tid + off];
        if (s2 > ss[tid] || (s2 == ss[tid] && i2 < si[tid])) {
          ss[tid] = s2; si[tid] = i2; sp[tid] = sp[tid + off];
        }
      }
      __syncthreads();
    }
    if (tid == 0) {
      if (sp[0] != 0xFFFFFFFFu) {
        out[sel]       = (float)(si[0] >> 14);     // / 16384
        out[256 + sel] = (float)(si[0] & 16383);   // % 16384
        out[512 + sel] = ss[0];
        cscore[sp[0]] = -INFINITY;                 // remove winner
      } else {
        out[sel] = 0.0f; out[256 + sel] = 0.0f; out[512 + sel] = 0.0f;
      }
    }
    __syncthreads();
  }
}

extern "C" void kernel_launch(void* const* d_in, const int* in_sizes, int n_in,
                              void* d_out, int out_size, void* d_ws, size_t ws_size,
                              hipStream_t stream)
{
  const float* ref = (const float*)d_in[0];
  const float* src = (const float*)d_in[1];
  float* out = (float*)d_out;

  char* ws = (char*)d_ws;
  unsigned* ghist  = (unsigned*)(ws + 0);              // 4096 * 4 = 16 KB
  unsigned* ctrl   = (unsigned*)(ws + 16384);          // [0]=tau, [1]=count
  float*    cscore = (float*)   (ws + 32768);          // CAP * 4
  unsigned* cidx   = (unsigned*)(ws + 32768 + CAP*4);  // CAP * 4

  (void)hipMemsetAsync(ws, 0, 32768, stream);          // zero hist + ctrl each call

  dim3 grid(MSRC / NBLK, NREF / MBLK);                 // 256 x 128 blocks
  gemm_topk_pass<<<grid, 256, 0, stream>>>(ref, src, ghist, ctrl, ctrl + 1,
                                           cscore, cidx, /*mode=*/0);
  find_thresh<<<1, 32, 0, stream>>>(ghist, ctrl);
  gemm_topk_pass<<<grid, 256, 0, stream>>>(ref, src, ghist, ctrl, ctrl + 1,
                                           cscore, cidx, /*mode=*/1);
  final_select<<<1, 256, 0, stream>>>(ctrl, cscore, cidx, out);
}